// PlasticLSTMCell_27762668601815
// MI455X (gfx1250) — compile-verified
//
#include <hip/hip_runtime.h>
#include <hip/hip_bf16.h>
#include <math.h>

// Sizes from the reference
#define BATCH 32
#define DIN   512
#define HID   1024
#define G4    (4 * HID)     // 4096 gate columns

typedef float v2f __attribute__((ext_vector_type(2)));
typedef float v4f __attribute__((ext_vector_type(4)));
typedef float v8f __attribute__((ext_vector_type(8)));

__device__ __forceinline__ float sigmoidf_fast(float x) {
    return 1.0f / (1.0f + __expf(-x));
}

// ---------------------------------------------------------------------------
// Kernel 1: plastic[b,h] = sum_i h0[b,i] * alpha[i,h] * Hebb0[b,i,h]
// One thread per (b,h). h contiguous across lanes -> coalesced 128B rows.
// Hebb0 (134 MB) pulled through L2 with default RT hint so it stays resident
// (L2 = 192 MB) for the second pass (hebb_update).
// ---------------------------------------------------------------------------
__global__ void plastic_kernel(const float* __restrict__ h0,
                               const float* __restrict__ alpha,
                               const float* __restrict__ Hebb0,
                               float* __restrict__ plastic) {
    const int t = blockIdx.x * blockDim.x + threadIdx.x;   // 0..B*H-1
    const int b = t >> 10;            // / HID
    const int h = t & (HID - 1);

    const float* __restrict__ hrow = h0 + b * HID;
    const float* __restrict__ Hb   = Hebb0 + ((size_t)b * HID) * HID + h;
    const float* __restrict__ ap   = alpha + h;

    float acc = 0.0f;
#pragma unroll 4
    for (int i = 0; i < HID; ++i) {
        if ((i & 7) == 0 && i + 32 < HID) {
            // strided stream: prefetch a few rows ahead (global_prefetch_b8)
            __builtin_prefetch(&Hb[(size_t)(i + 32) * HID], 0, 1);
        }
        acc = fmaf(hrow[i] * ap[(size_t)i * HID], Hb[(size_t)i * HID], acc);
    }
    plastic[t] = acc;
}

// ---------------------------------------------------------------------------
// Kernel 2: gates[b,n] = bias[n] + h0[b,:] @ Wh[:,n] + x[b,:] @ Wx[:,n]
// fp32 WMMA: V_WMMA_F32_16X16X4_F32.  One wave owns one 16-wide column tile
// and BOTH 16-row batch tiles (two accumulators sharing the B fragment).
//
// fp32 16x16x4 fragment layout (ISA 7.12.2):
//   A (16x4): lanes 0-15 hold M=lane, {K=0 (vgpr0), K=1 (vgpr1)};
//             lanes 16-31 hold M=lane-16, {K=2, K=3}.
//   B (4x16): lanes 0-15 hold N=lane, {K=0, K=1}; lanes 16-31 {K=2, K=3}.
//   C/D (16x16, 8 vgprs): lanes 0-15: M=v, N=lane; lanes 16-31: M=v+8.
// ---------------------------------------------------------------------------
__global__ void gates_gemm_kernel(const float* __restrict__ h0,
                                  const float* __restrict__ x,
                                  const float* __restrict__ Wh,
                                  const float* __restrict__ Wx,
                                  const float* __restrict__ bias,
                                  float* __restrict__ gates) {
    const int lane  = threadIdx.x & 31;
    const int wave  = threadIdx.x >> 5;
    const int waves = blockDim.x >> 5;
    const int n0    = (blockIdx.x * waves + wave) * 16;   // column tile base
    const int nl    = lane & 15;                          // M (A) or N (B) in-tile
    const int kb    = (lane >> 4) * 2;                    // K sub-offset: 0 or 2

    v8f acc0 = {};   // batch rows 0..15
    v8f acc1 = {};   // batch rows 16..31

    // ---- h0 (B x 1024) @ Wh (1024 x 4096) ----
#pragma unroll 2
    for (int k0 = 0; k0 < HID; k0 += 4) {
        v2f a0 = *(const v2f*)(h0 + nl * HID + k0 + kb);
        v2f a1 = *(const v2f*)(h0 + (nl + 16) * HID + k0 + kb);
        v2f bb;
        bb.x = Wh[(size_t)(k0 + kb) * G4 + n0 + nl];
        bb.y = Wh[(size_t)(k0 + kb + 1) * G4 + n0 + nl];
        acc0 = __builtin_amdgcn_wmma_f32_16x16x4_f32(false, a0, false, bb,
                                                     (short)0, acc0, false, false);
        acc1 = __builtin_amdgcn_wmma_f32_16x16x4_f32(false, a1, false, bb,
                                                     (short)0, acc1, false, false);
    }

    // ---- x (B x 512) @ Wx (512 x 4096) ----
#pragma unroll 2
    for (int k0 = 0; k0 < DIN; k0 += 4) {
        v2f a0 = *(const v2f*)(x + nl * DIN + k0 + kb);
        v2f a1 = *(const v2f*)(x + (nl + 16) * DIN + k0 + kb);
        v2f bb;
        bb.x = Wx[(size_t)(k0 + kb) * G4 + n0 + nl];
        bb.y = Wx[(size_t)(k0 + kb + 1) * G4 + n0 + nl];
        acc0 = __builtin_amdgcn_wmma_f32_16x16x4_f32(false, a0, false, bb,
                                                     (short)0, acc0, false, false);
        acc1 = __builtin_amdgcn_wmma_f32_16x16x4_f32(false, a1, false, bb,
                                                     (short)0, acc1, false, false);
    }

    // ---- epilogue: add bias, scatter C tiles to gates workspace ----
    const float bia   = bias[n0 + nl];
    const int   mhalf = (lane >> 4) * 8;
#pragma unroll
    for (int v = 0; v < 8; ++v) {
        const int row = v + mhalf;
        gates[(size_t)row * G4 + n0 + nl]        = acc0[v] + bia;
        gates[(size_t)(row + 16) * G4 + n0 + nl] = acc1[v] + bia;
    }
}

// ---------------------------------------------------------------------------
// Kernel 3: LSTM pointwise. gates layout [f | i | o | g] along columns.
// g gets + plastic.  Writes h1, c1 (outputs) and tanh_g (workspace).
// ---------------------------------------------------------------------------
__global__ void pointwise_kernel(const float* __restrict__ gates,
                                 const float* __restrict__ plastic,
                                 const float* __restrict__ c0,
                                 float* __restrict__ h1,
                                 float* __restrict__ c1,
                                 float* __restrict__ tg) {
    const int t = blockIdx.x * blockDim.x + threadIdx.x;   // 0..B*H-1
    const int b = t >> 10;
    const int h = t & (HID - 1);
    const float* __restrict__ g = gates + (size_t)b * G4;

    const float f_g = g[h];
    const float i_g = g[h + HID];
    const float o_g = g[h + 2 * HID];
    const float g_g = g[h + 3 * HID] + plastic[t];

    const float tgv = tanhf(g_g);
    const float c   = sigmoidf_fast(f_g) * c0[t] + sigmoidf_fast(i_g) * tgv;
    c1[t] = c;
    h1[t] = sigmoidf_fast(o_g) * tanhf(c);
    tg[t] = tgv;
}

// ---------------------------------------------------------------------------
// Kernel 4: Hebb1[b,i,h] = clip(Hebb0[b,i,h] + eta*h0[b,i]*tanh_g[b,h], -1, 1)
// float4 streaming RMW over 33.5M elements. Hebb0 loads hit L2 (warmed by
// kernel 1); Hebb1 stores are NON-TEMPORAL so the 134 MB output stream does
// not evict Hebb0 from the 192 MB L2 while we are still reading it.
// ---------------------------------------------------------------------------
__global__ void hebb_update_kernel(const float* __restrict__ Hebb0,
                                   const float* __restrict__ h0,
                                   const float* __restrict__ tg,
                                   const float* __restrict__ eta,
                                   float* __restrict__ Hebb1) {
    const size_t v = (size_t)blockIdx.x * blockDim.x + threadIdx.x; // float4 index
    const int    h4   = (int)(v & 255);          // HID/4 = 256 float4 per row
    const size_t rest = v >> 8;
    const int    i    = (int)(rest & (HID - 1));
    const int    b    = (int)(rest >> 10);

    const float scale = eta[0] * h0[b * HID + i];

    const v4f hv = *(const v4f*)(Hebb0 + v * 4);
    const v4f tv = *(const v4f*)(tg + ((size_t)b * HID) + h4 * 4);

    v4f r;
#pragma unroll
    for (int c = 0; c < 4; ++c) {
        float z = fmaf(scale, tv[c], hv[c]);
        z = fminf(1.0f, fmaxf(-1.0f, z));
        r[c] = z;
    }
    __builtin_nontemporal_store(r, (v4f*)(Hebb1 + v * 4));
}

// ---------------------------------------------------------------------------
// Launch
// ---------------------------------------------------------------------------
extern "C" void kernel_launch(void* const* d_in, const int* in_sizes, int n_in,
                              void* d_out, int out_size, void* d_ws, size_t ws_size,
                              hipStream_t stream) {
    const float* x     = (const float*)d_in[0];
    const float* h0    = (const float*)d_in[1];
    const float* c0    = (const float*)d_in[2];
    const float* Hebb0 = (const float*)d_in[3];
    const float* Wh    = (const float*)d_in[4];
    const float* Wx    = (const float*)d_in[5];
    const float* bias  = (const float*)d_in[6];
    const float* alpha = (const float*)d_in[7];
    const float* eta   = (const float*)d_in[8];

    float* out   = (float*)d_out;
    float* h1    = out;                          // B*H
    float* c1    = out + BATCH * HID;            // B*H
    float* Hebb1 = out + 2 * BATCH * HID;        // B*H*H

    float* ws      = (float*)d_ws;
    float* plastic = ws;                              // B*H      = 32768 f
    float* gates   = ws + BATCH * HID;                // B*4H     = 131072 f
    float* tg      = ws + BATCH * HID + BATCH * G4;   // B*H      = 32768 f

    // 1) plastic reduction (warms L2 with Hebb0)
    plastic_kernel<<<(BATCH * HID) / 256, 256, 0, stream>>>(h0, alpha, Hebb0, plastic);

    // 2) gates GEMM via fp32 WMMA: 256 column-tiles, 8 waves/block -> 32 blocks
    gates_gemm_kernel<<<G4 / (16 * 8), 256, 0, stream>>>(h0, x, Wh, Wx, bias, gates);

    // 3) pointwise LSTM
    pointwise_kernel<<<(BATCH * HID) / 256, 256, 0, stream>>>(gates, plastic, c0,
                                                              h1, c1, tg);

    // 4) Hebb update, float4-vectorized: B*H*H/4 = 8,388,608 threads
    hebb_update_kernel<<<(BATCH * HID * (HID / 4)) / 256, 256, 0, stream>>>(
        Hebb0, h0, tg, eta, Hebb1);
}